// ReConv_torch_20246475833696
// MI455X (gfx1250) — compile-verified
//
#include <hip/hip_runtime.h>
#include <stdint.h>

// ---------------- problem constants (match reference) ----------------
#define BN    4
#define CIN   64
#define HH    32
#define WW    32
#define KK    3
#define PADN  1
#define OC    64
#define LMAX  144
#define HP    34            // H + 2*PAD
#define WP    34            // W + 2*PAD
#define HOUT  32
#define WOUT  32
#define PP    (HOUT*WOUT)   // 1024
#define KDIM  (CIN*KK*KK)   // 576  (GEMM K)
#define NDIM  (BN*PP)       // 4096 (GEMM N)

typedef __attribute__((ext_vector_type(2))) float        v2f;
typedef __attribute__((ext_vector_type(8))) float        v8f;
typedef __attribute__((ext_vector_type(4))) unsigned int u32x4;
typedef __attribute__((ext_vector_type(4))) int          i32x4;
typedef __attribute__((ext_vector_type(8))) int          i32x8;

// LDS byte offset of a __shared__ pointer (generic -> addrspace(3) -> int)
__device__ __forceinline__ unsigned lds_byte_offset(const void* p) {
  return (unsigned)(unsigned long long)
      (const __attribute__((address_space(3))) char*)p;
}

// ---------------------------------------------------------------------
// Kernel 1: deterministic scatter-sum of sparse taps into a dense 3x3
// weight tensor Wd[OC][CIN*3*3], plus bias_full[OC] from bias_index.
// One thread per dense slot scans all taps -> no atomics, bitwise
// deterministic across replays.
// ---------------------------------------------------------------------
__global__ void build_weights(const int* __restrict__ tap_idx,
                              const float* __restrict__ tap_w,
                              const int* __restrict__ bias_index,
                              const float* __restrict__ bias_value,
                              float* __restrict__ wd,
                              float* __restrict__ bias_full) {
  int idx = blockIdx.x * blockDim.x + threadIdx.x;
  if (idx < OC * KDIM) {
    int o = idx / KDIM;
    int kcol = idx - o * KDIM;
    float s = 0.f;
    for (int l = 0; l < LMAX; ++l) {
      int t  = tap_idx[o * LMAX + l];        // c*HP*WP + kh*WP + kw
      int c  = t / (HP * WP);
      int r  = t - c * (HP * WP);
      int kh = r / WP;
      int kw = r - kh * WP;
      int col = (c * KK + kh) * KK + kw;     // dense K index
      if (col == kcol) s += tap_w[o * LMAX + l];
    }
    wd[idx] = s;
  } else if (idx < OC * KDIM + OC) {
    int j = idx - OC * KDIM;
    float s = 0.f;
    for (int i = 0; i < OC; ++i)
      if (bias_index[i] == j) s += bias_value[i];
    bias_full[j] = s;
  }
}

// ---------------------------------------------------------------------
// Kernel 2: materialize zero-padded images imgp[B][CIN][34][34] in ws.
// ---------------------------------------------------------------------
__global__ void pad_images(const float* __restrict__ images,
                           float* __restrict__ imgp) {
  int idx = blockIdx.x * blockDim.x + threadIdx.x;
  if (idx >= BN * CIN * HP * WP) return;
  int x  = idx % WP;
  int t  = idx / WP;
  int y  = t % HP;
  int bc = t / HP;
  float v = 0.f;
  if (x >= PADN && x <= WW && y >= PADN && y <= HH)
    v = images[bc * (HH * WW) + (y - PADN) * WW + (x - PADN)];
  imgp[idx] = v;
}

// ---------------------------------------------------------------------
// Kernel 3: implicit-GEMM 3x3 conv on V_WMMA_F32_16X16X4_F32.
// Block = 256 threads = 8 wave32s; computes C tile M=64 (all OC) x N=64
// (64 consecutive output pixels of one batch -> two image rows).
// Wave 0 DMAs the image slab [64ch][4 padded rows][34 cols] into LDS via
// the Tensor Data Mover, then every wave (mi 0..3, ni 0..1) runs the
// K=576 loop with two 16x16 f32 accumulators.
// ---------------------------------------------------------------------
__global__ __launch_bounds__(256, 1)
void reconv_wmma(const float* __restrict__ wd,
                 const float* __restrict__ bias,
                 const float* __restrict__ imgp,
                 float* __restrict__ out) {
  // [c][row][x] : row stride 34 floats, c stride 136 floats (34,816 B)
  __shared__ __align__(16) float smem[CIN * 4 * WP];

  const int tid    = threadIdx.x;
  const int wave   = tid >> 5;
  const int lane   = tid & 31;
  const int lane15 = lane & 15;
  const int khi    = (lane >> 4) << 1;     // K sub-index pair: 0 or 2
  const int mi     = wave >> 1;            // 0..3 : M tile (16 rows each)
  const int ni     = wave & 1;             // 0..1 : N half (32 cols each)

  const int n0   = blockIdx.x * 64;        // GEMM N base
  const int bimg = n0 >> 10;               // batch
  const int p0   = n0 & (PP - 1);          // first output pixel
  const int h0   = p0 >> 5;                // padded-row base (rows h0..h0+3)

  // ---- TDM async tile load: global -> LDS (wave 0 only; per-wave cnt) ----
  if (wave == 0) {
    unsigned lds = lds_byte_offset(smem);
    unsigned long long ga = (unsigned long long)(const void*)
        (imgp + (size_t)bimg * CIN * HP * WP + (size_t)h0 * WP);
    u32x4 g0; i32x8 g1; i32x4 g2, g3;
    // D# group0: count=1 | lds_addr | global_addr | type=2
    g0[0] = 1u;
    g0[1] = lds;
    g0[2] = (unsigned)(ga & 0xFFFFFFFFull);
    g0[3] = (unsigned)((ga >> 32) & 0x1FFFFFFull) | (2u << 30);
    // D# group1: data_size=4B; dims: x=34, y=4, z=64; strides 34 / 1156
    g1[0] = (int)(2u << 16);                 // data_size = 2 (4 bytes)
    g1[1] = (int)((unsigned)WP << 16);       // tensor_dim0 = 34
    g1[2] = (int)(4u << 16);                 // tensor_dim1 = 4
    g1[3] = (int)((unsigned)WP << 16);       // tile_dim0   = 34
    g1[4] = (int)((64u << 16) | 4u);         // tile_dim2=64 | tile_dim1=4
    g1[5] = (int)WP;                         // tensor_dim0_stride = 34
    g1[6] = (int)(((unsigned)(HP * WP) & 0xFFFFu) << 16); // dim1_stride lo16
    g1[7] = (int)((unsigned)(HP * WP) >> 16);             // dim1_stride hi
    // D# group2: tensor_dim2=64, tensor_dim2_stride=1156, tile_dim3=0
    g2[0] = CIN; g2[1] = 0; g2[2] = HP * WP; g2[3] = 0;
    g3[0] = 0; g3[1] = 0; g3[2] = 0; g3[3] = 0;
#if defined(__clang_major__) && (__clang_major__ >= 23)
    i32x8 g4 = {0, 0, 0, 0, 0, 0, 0, 0};
    __builtin_amdgcn_tensor_load_to_lds(g0, g1, g2, g3, g4, 0);
#else
    __builtin_amdgcn_tensor_load_to_lds(g0, g1, g2, g3, 0);
#endif
    __builtin_amdgcn_s_wait_tensorcnt(0);
  }
  __syncthreads();

  // ---- K loop: 144 steps of K=4, 2 WMMAs per step ----
  const int m16  = mi * 16;
  const int arow = (m16 + lane15) * KDIM;  // this lane's A-matrix row
  __builtin_prefetch(wd + arow, 0, 3);     // global_prefetch_b8 (L2-hot A)

  v8f acc0 = {0.f, 0.f, 0.f, 0.f, 0.f, 0.f, 0.f, 0.f};
  v8f acc1 = {0.f, 0.f, 0.f, 0.f, 0.f, 0.f, 0.f, 0.f};
  const int rowbase = ni * WP + lane15;    // ni selects image row 0/1

#pragma unroll 4
  for (int k0 = 0; k0 < KDIM; k0 += 4) {
    // A fragment: lanes 0-15 hold K=k0,k0+1; lanes 16-31 hold K=k0+2,k0+3
    v2f a = *(const v2f*)(wd + arow + k0 + khi);

    int k = k0 + khi;
    int c = k / 9, r = k - 9 * c, kh = r / 3, kw = r - 3 * kh;
    int bidx = c * (4 * WP) + kh * WP + kw + rowbase;
    v2f b0, b1;
    b0.x = smem[bidx];
    b1.x = smem[bidx + 16];

    k += 1;
    c = k / 9; r = k - 9 * c; kh = r / 3; kw = r - 3 * kh;
    bidx = c * (4 * WP) + kh * WP + kw + rowbase;
    b0.y = smem[bidx];
    b1.y = smem[bidx + 16];

    acc0 = __builtin_amdgcn_wmma_f32_16x16x4_f32(
        false, a, false, b0, (short)0, acc0, false, false);
    acc1 = __builtin_amdgcn_wmma_f32_16x16x4_f32(
        false, a, false, b1, (short)0, acc1, false, false);
  }

  // ---- epilogue: bias + store (C layout: VGPR r -> M=r / r+8) ----
  const int ncol0 = ni * 32 + lane15;
  const int ncol1 = ncol0 + 16;
  const int ph    = (lane >> 4) << 3;      // upper half-lanes hold M+8
#pragma unroll
  for (int r2 = 0; r2 < 8; ++r2) {
    int oc   = m16 + ph + r2;
    float bv = bias[oc];
    size_t base = ((size_t)(bimg * OC + oc) << 10) + (size_t)p0;
    out[base + ncol0] = acc0[r2] + bv;
    out[base + ncol1] = acc1[r2] + bv;
  }
}

// ---------------------------------------------------------------------
extern "C" void kernel_launch(void* const* d_in, const int* in_sizes, int n_in,
                              void* d_out, int out_size, void* d_ws, size_t ws_size,
                              hipStream_t stream) {
  (void)in_sizes; (void)n_in; (void)out_size; (void)ws_size;
  const float* images     = (const float*)d_in[0];
  const int*   tap_idx    = (const int*)d_in[1];
  const float* tap_w      = (const float*)d_in[2];
  const int*   bias_index = (const int*)d_in[3];
  const float* bias_value = (const float*)d_in[4];
  float*       out        = (float*)d_out;

  // workspace layout: Wd[64][576] | bias_full[64] | imgp[4][64][34][34]
  float* wd        = (float*)d_ws;
  float* bias_full = wd + OC * KDIM;
  float* imgp      = bias_full + OC;       // offset 147,712 B (16B aligned)

  int n1 = OC * KDIM + OC;                 // 36,928 threads
  build_weights<<<(n1 + 255) / 256, 256, 0, stream>>>(
      tap_idx, tap_w, bias_index, bias_value, wd, bias_full);

  int n2 = BN * CIN * HP * WP;             // 295,936 threads
  pad_images<<<(n2 + 255) / 256, 256, 0, stream>>>(images, imgp);

  reconv_wmma<<<NDIM / 64, 256, 0, stream>>>(wd, bias_full, imgp, out);
}